// YOLOv3Loss_89189290868800
// MI455X (gfx1250) — compile-verified
//
#include <hip/hip_runtime.h>

typedef __attribute__((ext_vector_type(2))) float v2f;
typedef __attribute__((ext_vector_type(8))) float v8f;

#define NCLS        25
#define NCH         30          // 5 + NCLS
#define NTGT        2048

// Cells per scale: B(64) * 3 anchors * g*g
#define CELLS0      32448       // g=13
#define CELLS1      129792      // g=26
#define CELLS2      519168      // g=52
#define OFF1        32448
#define OFF2        162240
#define TOTAL_CELLS 681408
#define MAIN_BLOCK  256
#define NB_MAIN     ((TOTAL_CELLS + MAIN_BLOCK - 1) / MAIN_BLOCK)   // 2662

// ---------- math helpers ----------
__device__ __forceinline__ float softplusf_(float x) {
    // stable softplus; bce(x,t) = softplus(x) - t*x
    return log1pf(expf(-fabsf(x))) + fmaxf(x, 0.0f);
}

__device__ __forceinline__ void load_sa(int scale, float sa[3][2]) {
    if (scale == 0) {
        sa[0][0]=116.0f/13.0f; sa[0][1]= 90.0f/13.0f;
        sa[1][0]=156.0f/13.0f; sa[1][1]=198.0f/13.0f;
        sa[2][0]=373.0f/13.0f; sa[2][1]=326.0f/13.0f;
    } else if (scale == 1) {
        sa[0][0]= 30.0f/26.0f; sa[0][1]= 61.0f/26.0f;
        sa[1][0]= 62.0f/26.0f; sa[1][1]= 45.0f/26.0f;
        sa[2][0]= 59.0f/26.0f; sa[2][1]=119.0f/26.0f;
    } else {
        sa[0][0]= 10.0f/52.0f; sa[0][1]= 13.0f/52.0f;
        sa[1][0]= 16.0f/52.0f; sa[1][1]= 30.0f/52.0f;
        sa[2][0]= 33.0f/52.0f; sa[2][1]= 23.0f/52.0f;
    }
}

__device__ __forceinline__ int best_anchor(float tw, float th, const float sa[3][2]) {
    int best = 0; float bi = -1.0f;
    #pragma unroll
    for (int j = 0; j < 3; ++j) {
        float inter = fminf(tw, sa[j][0]) * fminf(th, sa[j][1]);
        float uni   = tw * th + sa[j][0] * sa[j][1] - inter;
        float iou   = inter / (uni + 1e-9f);
        if (iou > bi) { bi = iou; best = j; }   // first max wins, like jnp.argmax
    }
    return best;
}

// ---------- WMMA-based 32-lane reduction (V_WMMA_F32_16X16X4_F32) ----------
// A = ones(16x4), B carries 64 values (2 per lane); D[m,n] = sum_k B[k,n].
// Since A is all-ones, any bijective packing of values into B is a valid
// partial reduction; 4 xor-shuffles finish the 16 column sums.
// EXEC must be all-ones at the call site (uniform control flow only).
__device__ __forceinline__ float wave_sum32(float v) {
    v2f a; a[0] = 1.0f; a[1] = 1.0f;
    v2f b; b[0] = v;    b[1] = 0.0f;
    v8f c = {};
    c = __builtin_amdgcn_wmma_f32_16x16x4_f32(false, a, false, b, (short)0, c, false, false);
    float s = c[0];                 // = v(lane%16) + v(lane%16 + 16), all lanes
    s += __shfl_xor(s, 1, 32);
    s += __shfl_xor(s, 2, 32);
    s += __shfl_xor(s, 4, 32);
    s += __shfl_xor(s, 8, 32);
    return s;                       // full 32-lane sum, broadcast to all lanes
}

// ---------- kernel 1: init winner table ----------
__global__ void yolo_init(int* __restrict__ win) {
    int i = blockIdx.x * blockDim.x + threadIdx.x;
    if (i < TOTAL_CELLS) win[i] = -1;
}

// ---------- kernel 2: scatter targets -> winner cells (deterministic: max target idx wins) ----------
__global__ void yolo_scatter(const float* __restrict__ tg, int* __restrict__ win) {
    int id = blockIdx.x * blockDim.x + threadIdx.x;
    if (id >= 3 * NTGT) return;
    int scale = id / NTGT;
    int t     = id - scale * NTGT;
    const float* T = tg + 6 * t;
    int g   = (scale == 0) ? 13 : (scale == 1) ? 26 : 52;
    int off = (scale == 0) ? 0  : (scale == 1) ? OFF1 : OFF2;
    float gf  = (float)g;
    float tx_ = T[2] * gf, ty_ = T[3] * gf;
    float tw_ = T[4] * gf, th_ = T[5] * gf;
    int gx = (int)floorf(tx_), gy = (int)floorf(ty_);
    if (gx < 0 || gx >= g || gy < 0 || gy >= g) return;   // OOB scatter dropped (JAX drop mode)
    float sa[3][2]; load_sa(scale, sa);
    int best = best_anchor(tw_, th_, sa);
    int b = (int)T[0];
    int cell = ((b * 3 + best) * g + gy) * g + gx;
    atomicMax(&win[off + cell], t);
}

// ---------- kernel 3: per-cell losses + WMMA block reduction ----------
__global__ void __launch_bounds__(MAIN_BLOCK)
yolo_main(const float* __restrict__ p0, const float* __restrict__ p1,
          const float* __restrict__ p2, const float* __restrict__ tg,
          const int* __restrict__ win, float* __restrict__ partials) {
    int tid = blockIdx.x * blockDim.x + threadIdx.x;
    float boxA = 0.0f, objA = 0.0f, clsA = 0.0f;

    if (tid < TOTAL_CELLS) {            // guard only; no early return (WMMA needs full EXEC)
        int scale, local, g;
        const float* pred;
        if (tid < OFF1)      { scale = 0; local = tid;        g = 13; pred = p0; }
        else if (tid < OFF2) { scale = 1; local = tid - OFF1; g = 26; pred = p1; }
        else                 { scale = 2; local = tid - OFF2; g = 52; pred = p2; }

        const float* pb = pred + (size_t)local * NCH;
        float po = pb[4];
        int w = win[tid];
        if (w < 0) {
            objA = 0.5f * softplusf_(po);                    // LAMBDA_NOOBJ * bce(po, 0)
        } else {
            objA = softplusf_(-po);                          // bce(po, 1) = softplus(po) - po
            const float* T = tg + 6 * w;
            float gf  = (float)g;
            int   cid = (int)T[1];
            float tx_ = T[2] * gf, ty_ = T[3] * gf;
            float tw_ = T[4] * gf, th_ = T[5] * gf;
            float sa[3][2]; load_sa(scale, sa);
            int best = best_anchor(tw_, th_, sa);
            float ttx = tx_ - floorf(tx_);
            float tty = ty_ - floorf(ty_);
            float ttw = logf(tw_ / sa[best][0] + 1e-16f);
            float tth = logf(th_ / sa[best][1] + 1e-16f);
            float px = pb[0], py = pb[1], pw = pb[2], ph = pb[3];
            float dx = pw - ttw, dy = ph - tth;
            boxA = (softplusf_(px) - ttx * px)
                 + (softplusf_(py) - tty * py)
                 + dx * dx + dy * dy;
            float cl = 0.0f;
            #pragma unroll
            for (int c = 0; c < NCLS; ++c) {
                float v = pb[5 + c];
                cl += softplusf_(v);                         // bce(v, onehot) summed
                if (c == cid) cl -= v;
            }
            clsA = cl;
        }
    }

    // per-wave reductions through the matrix pipe (uniform control flow here)
    float bS = wave_sum32(boxA);
    float oS = wave_sum32(objA);
    float cS = wave_sum32(clsA);

    __shared__ float sm[MAIN_BLOCK / 32][3];
    int wid  = threadIdx.x >> 5;
    int lane = threadIdx.x & 31;
    if (lane == 0) { sm[wid][0] = bS; sm[wid][1] = oS; sm[wid][2] = cS; }
    __syncthreads();
    if (threadIdx.x == 0) {
        float b = 0.0f, o = 0.0f, c = 0.0f;
        #pragma unroll
        for (int i = 0; i < MAIN_BLOCK / 32; ++i) { b += sm[i][0]; o += sm[i][1]; c += sm[i][2]; }
        partials[blockIdx.x]               = 5.0f * b;       // LAMBDA_COORD
        partials[NB_MAIN + blockIdx.x]     = o;
        partials[2 * NB_MAIN + blockIdx.x] = c;
    }
}

// ---------- kernel 4: single-wave WMMA-accumulated final reduce ----------
__global__ void yolo_reduce(const float* __restrict__ partials, float* __restrict__ out) {
    int lane = threadIdx.x;              // exactly 32 threads
    float totals[3];
    v2f a; a[0] = 1.0f; a[1] = 1.0f;     // ones A, reused each iteration
    #pragma unroll
    for (int q = 0; q < 3; ++q) {
        const float* p = partials + q * NB_MAIN;
        v8f c = {};
        for (int base = 0; base < NB_MAIN; base += 64) {
            int i0 = base + lane, i1 = base + 32 + lane;
            v2f b;
            b[0] = (i0 < NB_MAIN) ? p[i0] : 0.0f;
            b[1] = (i1 < NB_MAIN) ? p[i1] : 0.0f;
            c = __builtin_amdgcn_wmma_f32_16x16x4_f32(false, a, false, b, (short)0, c, false, false);
        }
        float s = c[0];
        s += __shfl_xor(s, 1, 32);
        s += __shfl_xor(s, 2, 32);
        s += __shfl_xor(s, 4, 32);
        s += __shfl_xor(s, 8, 32);
        totals[q] = s;
    }
    if (lane == 0) {
        out[0] = totals[0] + totals[1] + totals[2];   // total
        out[1] = totals[0];                            // box
        out[2] = totals[1];                            // obj
        out[3] = totals[2];                            // cls
    }
}

extern "C" void kernel_launch(void* const* d_in, const int* in_sizes, int n_in,
                              void* d_out, int out_size, void* d_ws, size_t ws_size,
                              hipStream_t stream) {
    const float* p0 = (const float*)d_in[0];   // pred_large  [64,3,13,13,30]
    const float* p1 = (const float*)d_in[1];   // pred_medium [64,3,26,26,30]
    const float* p2 = (const float*)d_in[2];   // pred_small  [64,3,52,52,30]
    const float* tg = (const float*)d_in[3];   // targets     [2048,6]
    (void)in_sizes; (void)n_in; (void)out_size; (void)ws_size;

    int*   win      = (int*)d_ws;                                   // 681408 ints
    size_t winBytes = ((size_t)TOTAL_CELLS * sizeof(int) + 255) & ~(size_t)255;
    float* partials = (float*)((char*)d_ws + winBytes);             // 3 * NB_MAIN floats

    yolo_init   <<<(TOTAL_CELLS + 255) / 256, 256, 0, stream>>>(win);
    yolo_scatter<<<(3 * NTGT + 255) / 256,   256, 0, stream>>>(tg, win);
    yolo_main   <<<NB_MAIN, MAIN_BLOCK, 0, stream>>>(p0, p1, p2, tg, win, partials);
    yolo_reduce <<<1, 32, 0, stream>>>(partials, (float*)d_out);
}